// BaseTransformStandalone_16054587753036
// MI455X (gfx1250) — compile-verified
//
#include <hip/hip_runtime.h>
#include <hip/hip_bf16.h>

// Problem constants (from reference)
#define B_   2
#define N_   6
#define D_   59
#define FH_  16
#define FW_  44
#define C_   80
#define X_   256
#define Y_   256
#define NPTS (D_ * FH_ * FW_)        // 41536 points per (b,n)  (divisible by 32)
#define PLANE (Y_ * X_)              // 65536
#define OUT_PER_B (C_ * PLANE)

typedef float v2f __attribute__((ext_vector_type(2)));
typedef float v8f __attribute__((ext_vector_type(8)));

#if defined(__gfx1250__) && __has_builtin(__builtin_amdgcn_wmma_f32_16x16x4_f32)
#define USE_WMMA4 1
#else
#define USE_WMMA4 0
#endif

// ---------------------------------------------------------------- helpers
__device__ __forceinline__ void inv3(const float* a, float* r) {
  float det = a[0] * (a[4] * a[8] - a[5] * a[7])
            - a[1] * (a[3] * a[8] - a[5] * a[6])
            + a[2] * (a[3] * a[7] - a[4] * a[6]);
  float id = 1.0f / det;
  r[0] = (a[4] * a[8] - a[5] * a[7]) * id;
  r[1] = (a[2] * a[7] - a[1] * a[8]) * id;
  r[2] = (a[1] * a[5] - a[2] * a[4]) * id;
  r[3] = (a[5] * a[6] - a[3] * a[8]) * id;
  r[4] = (a[0] * a[8] - a[2] * a[6]) * id;
  r[5] = (a[2] * a[3] - a[0] * a[5]) * id;
  r[6] = (a[3] * a[7] - a[4] * a[6]) * id;
  r[7] = (a[1] * a[6] - a[0] * a[7]) * id;
  r[8] = (a[0] * a[4] - a[1] * a[3]) * id;
}

__device__ __forceinline__ void mm3(const float* a, const float* b, float* r) {
#pragma unroll
  for (int i = 0; i < 3; ++i)
#pragma unroll
    for (int j = 0; j < 3; ++j)
      r[i * 3 + j] = a[i * 3 + 0] * b[0 + j] + a[i * 3 + 1] * b[3 + j] + a[i * 3 + 2] * b[6 + j];
}

// Voxelize one geometry point; returns output base index or -1 if culled.
// Matches reference: coords = int32 trunc of (g - (BX - DX/2)) / DX.
__device__ __forceinline__ int voxel_out(float gx, float gy, float gz, int b) {
  int cx = (int)((gx + 51.2f) / 0.4f);
  int cy = (int)((gy + 51.2f) / 0.4f);
  int cz = (int)((gz + 5.0f) / 8.0f);
  if (cx < 0 || cx >= X_ || cy < 0 || cy >= Y_ || cz != 0) return -1;
  return b * OUT_PER_B + cy * X_ + cx;   // out[b][c][cy][cx], channel added later
}

// ---------------------------------------------------------------- kernels
__global__ void zero_out_kernel(float4* __restrict__ o, int n4) {
  int i = blockIdx.x * blockDim.x + threadIdx.x;
  if (i < n4) o[i] = make_float4(0.f, 0.f, 0.f, 0.f);
}

// Per-(b,n) transform precompute: ws[bn*32 + ...] =
//   [0..8]  invPost (3x3)      [9..11]  post_trans
//   [12..20] M = invL2E*rots*invIntr   [21..23] t = invL2E*(trans - l2e_trans)
__global__ void precompute_tf_kernel(const float* __restrict__ rots,
                                     const float* __restrict__ trans,
                                     const float* __restrict__ intrins,
                                     const float* __restrict__ post_rots,
                                     const float* __restrict__ post_trans,
                                     const float* __restrict__ l2e_rots,
                                     const float* __restrict__ l2e_trans,
                                     float* __restrict__ ws) {
  int t = threadIdx.x;
  if (t >= B_ * N_) return;
  int b = t / N_;
  float R[9], K[9], P[9], L[9];
#pragma unroll
  for (int i = 0; i < 9; ++i) {
    R[i] = rots[t * 9 + i];
    K[i] = intrins[t * 9 + i];
    P[i] = post_rots[t * 9 + i];
    L[i] = l2e_rots[b * 9 + i];
  }
  float iK[9], iP[9], iL[9];
  inv3(K, iK); inv3(P, iP); inv3(L, iL);
  float RK[9], M[9];
  mm3(R, iK, RK);
  mm3(iL, RK, M);
  float tv[3], tt[3];
#pragma unroll
  for (int i = 0; i < 3; ++i) tv[i] = trans[t * 3 + i] - l2e_trans[b * 3 + i];
#pragma unroll
  for (int i = 0; i < 3; ++i)
    tt[i] = iL[i * 3 + 0] * tv[0] + iL[i * 3 + 1] * tv[1] + iL[i * 3 + 2] * tv[2];

  float* w = ws + t * 32;
#pragma unroll
  for (int i = 0; i < 9; ++i) w[i] = iP[i];
#pragma unroll
  for (int i = 0; i < 3; ++i) w[9 + i] = post_trans[t * 3 + i];
#pragma unroll
  for (int i = 0; i < 9; ++i) w[12 + i] = M[i];
#pragma unroll
  for (int i = 0; i < 3; ++i) w[21 + i] = tt[i];
}

// Fused geometry + BEV scatter. Grid: (ceil(NPTS/256), N, B), block 256 (8 waves).
__global__ void bev_scatter_kernel(const float* __restrict__ x,
                                   const float* __restrict__ ws,
                                   float* __restrict__ out) {
  const int lane = threadIdx.x & 31;
  const int wave = threadIdx.x >> 5;
  const int n = blockIdx.y, b = blockIdx.z;
  const int bn = b * N_ + n;
  const int base = (blockIdx.x * 8 + wave) * 32;   // 32 points per wave
  if (base >= NPTS) return;

  const float* w = ws + bn * 32;
  const float iP0 = w[0], iP1 = w[1], iP2 = w[2];
  const float iP3 = w[3], iP4 = w[4], iP5 = w[5];
  const float iP6 = w[6], iP7 = w[7], iP8 = w[8];
  const float pt0 = w[9], pt1 = w[10], pt2 = w[11];

  // Each lane computes its own point's frustum ray + un-projected q.
  const int p   = base + lane;                 // NPTS % 32 == 0, always valid
  const int d_i = p / (FH_ * FW_);
  const int rem = p - d_i * (FH_ * FW_);
  const int h   = rem / FW_;
  const int wc  = rem - h * FW_;

  const float u   = (float)wc * (703.0f / 43.0f);   // linspace(0, IW-1, FW)
  const float v   = (float)h * 17.0f;               // linspace(0, IH-1, FH)
  const float dep = (float)d_i + 1.0f;              // arange(1, 60)

  const float ax = u - pt0, ay = v - pt1, az = dep - pt2;
  const float r0 = iP0 * ax + iP1 * ay + iP2 * az;
  const float r1 = iP3 * ax + iP4 * ay + iP5 * az;
  const float r2 = iP6 * ax + iP7 * ay + iP8 * az;
  const float q0 = r0 * r2, q1 = r1 * r2, q2 = r2;  // homogeneous coord q3 = 1

  int ob1, ob2;
#if USE_WMMA4
  // g = M*q + t as D = A(16x4) x B(4x16):
  //   A rows m<3: [M[m][0], M[m][1], M[m][2], t[m]], rows 3..15 zero (uniform per block)
  //   B cols n = points: [q0, q1, q2, 1]^T
  // D lanes 0..15 hold point `lane`: vgpr0=gx, vgpr1=gy, vgpr2=gz.
  const int m = lane & 15;
  float a0 = 0.0f, a1 = 0.0f;
  if (m < 3) {
    if (lane < 16) { a0 = w[12 + m * 3 + 0]; a1 = w[12 + m * 3 + 1]; }
    else           { a0 = w[12 + m * 3 + 2]; a1 = w[21 + m]; }
  }
  v2f A; A[0] = a0; A[1] = a1;

  const float q2lo = __shfl(q2, lane & 15, 32);     // q2 of point (lane-16) for hi lanes
  v2f Bm1; Bm1[0] = (lane < 16) ? q0 : q2lo;
           Bm1[1] = (lane < 16) ? q1 : 1.0f;
  v8f cz = {};
  v8f g1 = __builtin_amdgcn_wmma_f32_16x16x4_f32(
      false, A, false, Bm1, (short)0, cz, false, false);

  const float q0hi = __shfl(q0, lane | 16, 32);     // q of point (lane+16) for lo lanes
  const float q1hi = __shfl(q1, lane | 16, 32);
  v2f Bm2; Bm2[0] = (lane < 16) ? q0hi : q2;
           Bm2[1] = (lane < 16) ? q1hi : 1.0f;
  v8f g2 = __builtin_amdgcn_wmma_f32_16x16x4_f32(
      false, A, false, Bm2, (short)0, cz, false, false);

  ob1 = voxel_out(g1[0], g1[1], g1[2], b);          // point `lane`        (lanes 0..15)
  ob2 = voxel_out(g2[0], g2[1], g2[2], b);          // point `lane + 16`   (lanes 0..15)
#else
  // Scalar fallback: each lane transforms its own point.
  const float M0 = w[12], M1 = w[13], M2 = w[14];
  const float M3 = w[15], M4 = w[16], M5 = w[17];
  const float M6 = w[18], M7 = w[19], M8 = w[20];
  const float t0 = w[21], t1 = w[22], t2 = w[23];
  const float gx = M0 * q0 + M1 * q1 + M2 * q2 + t0;
  const float gy = M3 * q0 + M4 * q1 + M5 * q2 + t1;
  const float gz = M6 * q0 + M7 * q1 + M8 * q2 + t2;
  ob1 = voxel_out(gx, gy, gz, b);
  ob2 = 0;
#endif

  // Cooperative scatter: all 32 lanes handle the 80 channels of each point.
  const float* xb = x + (size_t)bn * (size_t)NPTS * C_;
  for (int j = 0; j < 32; ++j) {
#if USE_WMMA4
    const int ob = __shfl((j < 16) ? ob1 : ob2, j & 15, 32);
#else
    const int ob = __shfl(ob1, j, 32);
#endif
    if (ob < 0) continue;
    const float* feat = xb + (size_t)(base + j) * C_;  // 80 contiguous floats
    const float f0 = feat[lane];
    const float f1 = feat[lane + 32];
    atomicAdd(&out[ob + lane * PLANE], f0);
    atomicAdd(&out[ob + (lane + 32) * PLANE], f1);
    if (lane < 16) {
      const float f2 = feat[lane + 64];
      atomicAdd(&out[ob + (lane + 64) * PLANE], f2);
    }
  }
}

// ---------------------------------------------------------------- launch
extern "C" void kernel_launch(void* const* d_in, const int* in_sizes, int n_in,
                              void* d_out, int out_size, void* d_ws, size_t ws_size,
                              hipStream_t stream) {
  const float* x          = (const float*)d_in[0];
  const float* rots       = (const float*)d_in[1];
  const float* trans      = (const float*)d_in[2];
  const float* intrins    = (const float*)d_in[3];
  const float* post_rots  = (const float*)d_in[4];
  const float* post_trans = (const float*)d_in[5];
  const float* l2e_rots   = (const float*)d_in[6];
  const float* l2e_trans  = (const float*)d_in[7];
  float* out = (float*)d_out;
  float* ws  = (float*)d_ws;

  const int n4 = out_size / 4;   // out_size = 2*80*256*256, divisible by 4
  zero_out_kernel<<<(n4 + 255) / 256, 256, 0, stream>>>((float4*)out, n4);

  precompute_tf_kernel<<<1, 32, 0, stream>>>(rots, trans, intrins, post_rots,
                                             post_trans, l2e_rots, l2e_trans, ws);

  dim3 grid((NPTS + 255) / 256, N_, B_);
  bev_scatter_kernel<<<grid, 256, 0, stream>>>(x, ws, out);
}